// sub_channel_attention_53575422050715
// MI455X (gfx1250) — compile-verified
//
#include <hip/hip_runtime.h>
#include <hip/hip_bf16.h>
#include <stddef.h>

// ---------------------------------------------------------------------------
// MI455X (gfx1250) implementation.
//
// Frozen shapes: B=256, C=1024, S=14 (196 pix), C2=15, CAT=1551, O=C/2=512.
// Heavy work = two fp32 GEMMs (K=1024 and K=1551, M=512, N=196 per batch)
// plus 256K tiny 14x14 covariance matmuls -> mapped onto
// V_WMMA_F32_16X16X4_F32. Workload is HBM-bound (~135 GFLOP vs ~2 GB of fp32
// traffic -> ~85us at 23.3 TB/s), so fp32 WMMA matches reference numerics
// while the GEMM hides global->LDS movement behind the ASYNCcnt path
// (GLOBAL_LOAD_ASYNC_TO_LDS + s_wait_asynccnt, double-buffered LDS tiles).
//
// Workspace layout (floats), requires ~417.1 MB of d_ws:
//   y1    :   752,640   (256*15*196)
//   zbuf  : 25,690,112  (256*512*196)   z_pre -> z (in place)
//   covdw : 77,822,976  (256*1551*196)  first used as cov (51,380,224 floats),
//                                       then reused as depthwise output
//   stats :     2,048   (mean/rstd for the two BatchNorms)
// ---------------------------------------------------------------------------

typedef float v2f __attribute__((ext_vector_type(2)));
typedef float v8f __attribute__((ext_vector_type(8)));
typedef int   v4i_vec __attribute__((vector_size(16)));   // matches builtin param

#define Bsz   256
#define Cch   1024
#define SS    196          // 14*14
#define C2ch  15
#define CATch 1551
#define Och   512

#define AS_GLOBAL __attribute__((address_space(1)))
#define AS_LDS    __attribute__((address_space(3)))

#if __has_builtin(__builtin_amdgcn_global_load_async_to_lds_b32) && \
    __has_builtin(__builtin_amdgcn_global_load_async_to_lds_b128)
#define HAVE_ASYNC_LDS 1
#else
#define HAVE_ASYNC_LDS 0
#endif

#if HAVE_ASYNC_LDS
#if __has_builtin(__builtin_amdgcn_s_wait_asynccnt)
#define WAIT_ASYNC(n) __builtin_amdgcn_s_wait_asynccnt(n)
#else
#define WAIT_ASYNC(n) asm volatile("s_wait_asynccnt %0" ::"n"(n) : "memory")
#endif
#else
#define WAIT_ASYNC(n)
#endif

// ---------------------------------------------------------------------------
// Kernel 1: SE block on the 15 sub-channels of y.
// y (B,15,196) -> y1 = y * sigmoid(fc2(relu(fc1(avgpool(y)))))
// ---------------------------------------------------------------------------
__global__ __launch_bounds__(256) void se_scale_kernel(
    const float* __restrict__ y, const float* __restrict__ fc1,
    const float* __restrict__ fc2, float* __restrict__ y1) {
  const int b = blockIdx.x;
  const int tid = threadIdx.x;
  __shared__ float gsum[C2ch];
  __shared__ float hbuf[2];
  __shared__ float sig[C2ch];
  if (tid < C2ch) gsum[tid] = 0.f;
  __syncthreads();
  const float* yb = y + (size_t)b * C2ch * SS;
  for (int i = tid; i < C2ch * SS; i += 256)
    atomicAdd(&gsum[i / SS], yb[i]);
  __syncthreads();
  if (tid < 2) {  // r = C2//7 = 2
    float acc = 0.f;
    for (int c = 0; c < C2ch; ++c)
      acc += (gsum[c] * (1.f / (float)SS)) * fc1[tid * C2ch + c];
    hbuf[tid] = fmaxf(acc, 0.f);
  }
  __syncthreads();
  if (tid < C2ch) {
    float v = hbuf[0] * fc2[tid * 2 + 0] + hbuf[1] * fc2[tid * 2 + 1];
    sig[tid] = 1.f / (1.f + __expf(-v));
  }
  __syncthreads();
  float* y1b = y1 + (size_t)b * C2ch * SS;
  for (int i = tid; i < C2ch * SS; i += 256)
    y1b[i] = yb[i] * sig[i / SS];
}

// ---------------------------------------------------------------------------
// Kernel 2: depthwise 3x3 on x, then per-(b,c) row covariance of the 14x14
// map via WMMA. One wave per (b,c); XC zero-padded to 16x16 in LDS.
// cov = XC @ XC^T / 13. For this symmetric product the A-fragment (16x4) and
// B-fragment (4x16 of XC^T) have the same per-lane layout, so one fragment
// feeds both WMMA operands.
// ---------------------------------------------------------------------------
#define COV_WAVES 4
__global__ __launch_bounds__(COV_WAVES * 32) void dconv_cov_kernel(
    const float* __restrict__ x, const float* __restrict__ w,
    const float* __restrict__ bias, float* __restrict__ cov) {
  __shared__ float raw[COV_WAVES][256];
  __shared__ float xc[COV_WAVES][256];
  const int tid = threadIdx.x;
  const int lane = tid & 31;
  const int wv = tid >> 5;
  const int bc = blockIdx.x * COV_WAVES + wv;   // (b*C + c), exact grid
  const int c = bc & (Cch - 1);

  for (int i = lane; i < 256; i += 32) { raw[wv][i] = 0.f; xc[wv][i] = 0.f; }
  __syncthreads();

  const float* src = x + (size_t)bc * SS;
  for (int e = lane; e < SS; e += 32)
    raw[wv][(e / 14 + 1) * 16 + (e % 14 + 1)] = src[e];
  __syncthreads();

  const float w0 = w[c*9+0], w1 = w[c*9+1], w2 = w[c*9+2];
  const float w3 = w[c*9+3], w4 = w[c*9+4], w5 = w[c*9+5];
  const float w6 = w[c*9+6], w7 = w[c*9+7], w8 = w[c*9+8];
  const float bch = bias[c];
  for (int e = lane; e < SS; e += 32) {
    const int i = e / 14, j = e % 14;
    const float* r0 = &raw[wv][(i + 0) * 16 + j];
    const float* r1 = &raw[wv][(i + 1) * 16 + j];
    const float* r2 = &raw[wv][(i + 2) * 16 + j];
    float a = bch;
    a += w0 * r0[0] + w1 * r0[1] + w2 * r0[2];
    a += w3 * r1[0] + w4 * r1[1] + w5 * r1[2];
    a += w6 * r2[0] + w7 * r2[1] + w8 * r2[2];
    xc[wv][i * 16 + j] = a;
  }
  __syncthreads();

  // subtract per-row mean (np.cov: rows = variables, cols = observations)
  if (lane < 14) {
    float m = 0.f;
#pragma unroll
    for (int j = 0; j < 14; ++j) m += xc[wv][lane * 16 + j];
    m *= (1.f / 14.f);
#pragma unroll
    for (int j = 0; j < 14; ++j) xc[wv][lane * 16 + j] -= m;
  }
  __syncthreads();

  // C(16x16) = XC(16x16) @ XC^T via 4 WMMA f32 16x16x4 steps
  v8f acc = {};
  const int l = lane & 15;
  const int koff = (lane >> 4) << 1;   // lanes 16-31 hold K+2/K+3
#pragma unroll
  for (int kk = 0; kk < 16; kk += 4) {
    v2f a;
    a.x = xc[wv][l * 16 + kk + koff];
    a.y = xc[wv][l * 16 + kk + koff + 1];
    acc = __builtin_amdgcn_wmma_f32_16x16x4_f32(
        false, a, false, a, (short)0, acc, false, false);
  }

  const int col = l;
  const int rbase = (lane >> 4) << 3;
#pragma unroll
  for (int v = 0; v < 8; ++v) {
    const int row = v + rbase;
    if (row < 14 && col < 14)
      cov[(size_t)bc * SS + row * 14 + col] = acc[v] * (1.f / 13.f);
  }
}

// ---------------------------------------------------------------------------
// Kernel 3/7: fp32 WMMA GEMM for the 1x1 convs.
//   Cout[b,m,n] = sum_k A[m,k] * Bact[b,k,n] + bias[m],  n < 196
// Block = 8 waves (256 thr) computing a 64(M) x 32(N) tile. K-chunks of 32,
// double-buffered in LDS; full chunks are fetched with
// GLOBAL_LOAD_ASYNC_TO_LDS (ASYNCcnt), overlapped with the WMMAs on the
// other buffer; the (rare) partial K-tail falls back to staged stores.
// Each wave owns one 16x16 accumulator -> 8 v_wmma_f32_16x16x4_f32 / chunk.
// ---------------------------------------------------------------------------
#define BSTRIDE 36  // B tile row stride: keeps b128 LDS targets 16B aligned
__global__ __launch_bounds__(256) void wmma_gemm_kernel(
    const float* __restrict__ A, const float* __restrict__ Bact,
    const float* __restrict__ bias, float* __restrict__ Cout,
    int K, int O) {
  __shared__ float Als[2][64][33];
  __shared__ __align__(16) float Bls[2][32][BSTRIDE];
  __shared__ __align__(16) float dump[4];
  const int tid = threadIdx.x;
  const int lane = tid & 31;
  const int wv = tid >> 5;       // 0..7
  const int mi = wv >> 1;        // 0..3  -> M sub-tile
  const int ni = wv & 1;         // 0..1  -> N sub-tile
  const int Noff = blockIdx.x * 32;
  const int Moff = blockIdx.y * 64;
  const int b = blockIdx.z;

  v8f acc = {};
  const int l = lane & 15;
  const int koff = (lane >> 4) << 1;
  const int m0 = mi * 16, n0 = ni * 16;

  // async-copy lane assignments (one chunk = 9 async instructions per wave)
  const int am = tid >> 2;              // A row 0..63
  const int ak = (tid & 3) * 8;         // A k-octet
  const int bk = tid >> 3;              // B row 0..31
  const int bn = (tid & 7) * 4;         // B col-quad (196 % 4 == 0 -> all-or-nothing)
  const bool bvalid = (Noff + bn) < SS;

  // staged fallback loader: fills the whole tile, zero-filling OOB entries
  auto stage = [&](int buf, int k0) {
    for (int i = tid; i < 64 * 32; i += 256) {
      const int m = i >> 5, k = i & 31;
      const int kg = k0 + k;
      Als[buf][m][k] = (kg < K) ? A[(size_t)(Moff + m) * K + kg] : 0.f;
    }
    for (int i = tid; i < 32 * 32; i += 256) {
      const int k = i >> 5, n = i & 31;
      const int kg = k0 + k, ng = Noff + n;
      Bls[buf][k][n] = (kg < K && ng < SS)
                           ? Bact[((size_t)b * K + kg) * SS + ng] : 0.f;
    }
  };

#if HAVE_ASYNC_LDS
  auto issue_async = [&](int buf, int k0) {  // only for full chunks (k0+32<=K)
    const float* asrc = &A[(size_t)(Moff + am) * K + k0 + ak];
    float* adst = &Als[buf][am][ak];
#pragma unroll
    for (int j = 0; j < 8; ++j)
      __builtin_amdgcn_global_load_async_to_lds_b32(
          (AS_GLOBAL int*)(asrc + j), (AS_LDS int*)(adst + j), 0, 0);
    // constant 1 instruction even on the N edge: invalid quads go to `dump`
    const float* bsrc =
        bvalid ? &Bact[((size_t)b * K + k0 + bk) * SS + Noff + bn] : Bact;
    float* bdst = bvalid ? &Bls[buf][bk][bn] : dump;
    __builtin_amdgcn_global_load_async_to_lds_b128(
        (AS_GLOBAL v4i_vec*)bsrc, (AS_LDS v4i_vec*)bdst, 0, 0);
  };
  // N-edge: zero invalid columns once; async copies never touch them
  if (Noff + 32 > SS) {
    for (int i = tid; i < 2 * 32 * 32; i += 256) {
      const int bufz = i >> 10, k = (i >> 5) & 31, n = i & 31;
      if (Noff + n >= SS) Bls[bufz][k][n] = 0.f;
    }
    __syncthreads();
  }
#endif

  const int nchunks = (K + 31) / 32;
  // prologue: chunk 0 is always full (K >= 32)
#if HAVE_ASYNC_LDS
  issue_async(0, 0);
#else
  stage(0, 0);
#endif

  for (int ch = 0; ch < nchunks; ++ch) {
    const int cur = ch & 1;
    bool next_async = false;
    if (ch + 1 < nchunks) {
      const int k0n = (ch + 1) * 32;
#if HAVE_ASYNC_LDS
      if (k0n + 32 <= K) { issue_async(cur ^ 1, k0n); next_async = true; }
      else              { stage(cur ^ 1, k0n); }
#else
      stage(cur ^ 1, k0n);
      if (k0n + 32 < K)
        __builtin_prefetch(&Bact[((size_t)b * K + k0n + 32) * SS + Noff], 0, 0);
#endif
    }
    if (next_async) { WAIT_ASYNC(9); } else { WAIT_ASYNC(0); }
    __syncthreads();  // chunk `ch` resident in LDS for all waves
#pragma unroll
    for (int kk = 0; kk < 32; kk += 4) {
      v2f a, bf;
      a.x  = Als[cur][m0 + l][kk + koff];
      a.y  = Als[cur][m0 + l][kk + koff + 1];
      bf.x = Bls[cur][kk + koff][n0 + l];
      bf.y = Bls[cur][kk + koff + 1][n0 + l];
      acc = __builtin_amdgcn_wmma_f32_16x16x4_f32(
          false, a, false, bf, (short)0, acc, false, false);
    }
    __syncthreads();  // all waves done reading `cur` before it is refilled
  }

  const int n = Noff + n0 + l;
  const int rbase = (lane >> 4) << 3;
#pragma unroll
  for (int v = 0; v < 8; ++v) {
    const int m = Moff + m0 + v + rbase;
    if (n < SS)
      Cout[((size_t)b * O + m) * SS + n] = acc[v] + bias[m];
  }
}

// ---------------------------------------------------------------------------
// Kernel 4/8: training-mode BatchNorm stats: per-channel mean / rsqrt(var+eps)
// over (N,H,W) = 256*196 = 50176 values. One block per channel.
// ---------------------------------------------------------------------------
__global__ __launch_bounds__(256) void bn_stats_kernel(
    const float* __restrict__ X, float* __restrict__ mean,
    float* __restrict__ rstd) {
  const int o = blockIdx.x;
  const int tid = threadIdx.x;
  float s = 0.f, sq = 0.f;
  for (int i = tid; i < Bsz * SS; i += 256) {
    const int b = i / SS, e = i % SS;
    const float v = X[((size_t)b * Och + o) * SS + e];
    s += v; sq += v * v;
  }
  __shared__ float ss[256], s2[256];
  ss[tid] = s; s2[tid] = sq;
  __syncthreads();
  for (int st = 128; st > 0; st >>= 1) {
    if (tid < st) { ss[tid] += ss[tid + st]; s2[tid] += s2[tid + st]; }
    __syncthreads();
  }
  if (tid == 0) {
    const float m = ss[0] * (1.f / (float)(Bsz * SS));
    const float var = s2[0] * (1.f / (float)(Bsz * SS)) - m * m;
    mean[o] = m;
    rstd[o] = rsqrtf(var + 1e-5f);
  }
}

// Kernel 5/9: apply BN then sigmoid (mode=0) or relu (mode=1), in place.
__global__ __launch_bounds__(256) void bn_apply_kernel(
    float* __restrict__ X, const float* __restrict__ mean,
    const float* __restrict__ rstd, const float* __restrict__ gamma,
    const float* __restrict__ beta, int mode) {
  const size_t idx = (size_t)blockIdx.x * 256 + threadIdx.x;
  const int o = (int)((idx / SS) % Och);
  const float v = (X[idx] - mean[o]) * rstd[o] * gamma[o] + beta[o];
  X[idx] = mode ? fmaxf(v, 0.f) : 1.f / (1.f + __expf(-v));
}

// ---------------------------------------------------------------------------
// Kernel 6: depthwise 3x3 over the *virtual* concat [y1 | z | x] (1551 ch).
// One block per (b, ch); never materializes cat.
// ---------------------------------------------------------------------------
__global__ __launch_bounds__(256) void dw_cat_kernel(
    const float* __restrict__ y1, const float* __restrict__ z,
    const float* __restrict__ x, const float* __restrict__ w,
    const float* __restrict__ bias, float* __restrict__ outp) {
  const int bc = blockIdx.x;
  const int b = bc / CATch;
  const int ch = bc % CATch;
  __shared__ float raw[256];
  const int tid = threadIdx.x;
  raw[tid] = 0.f;
  __syncthreads();
  const float* src;
  if (ch < C2ch)            src = y1 + ((size_t)b * C2ch + ch) * SS;
  else if (ch < C2ch + Och) src = z  + ((size_t)b * Och + (ch - C2ch)) * SS;
  else                      src = x  + ((size_t)b * Cch + (ch - C2ch - Och)) * SS;
  if (tid < SS)
    raw[(tid / 14 + 1) * 16 + (tid % 14 + 1)] = src[tid];
  __syncthreads();
  if (tid < SS) {
    const int i = tid / 14, j = tid % 14;
    float acc = bias[ch];
#pragma unroll
    for (int di = 0; di < 3; ++di)
#pragma unroll
      for (int dj = 0; dj < 3; ++dj)
        acc += w[ch * 9 + di * 3 + dj] * raw[(i + di) * 16 + (j + dj)];
    outp[((size_t)b * CATch + ch) * SS + tid] = acc;
  }
}

// ---------------------------------------------------------------------------
extern "C" void kernel_launch(void* const* d_in, const int* in_sizes, int n_in,
                              void* d_out, int out_size, void* d_ws,
                              size_t ws_size, hipStream_t stream) {
  const float* y       = (const float*)d_in[0];
  const float* x       = (const float*)d_in[1];
  const float* fc1_w   = (const float*)d_in[2];
  const float* fc2_w   = (const float*)d_in[3];
  const float* conv_w  = (const float*)d_in[4];
  const float* conv_b  = (const float*)d_in[5];
  const float* dconv_w = (const float*)d_in[6];
  const float* dconv_b = (const float*)d_in[7];
  const float* dw1_w   = (const float*)d_in[8];
  const float* dw1_b   = (const float*)d_in[9];
  const float* pw1_w   = (const float*)d_in[10];
  const float* pw1_b   = (const float*)d_in[11];
  const float* gamma   = (const float*)d_in[12];
  const float* beta    = (const float*)d_in[13];
  // d_in[14] = n (== 0 at these frozen shapes; resize loop is a no-op)
  float* out = (float*)d_out;

  float* ws    = (float*)d_ws;
  float* y1    = ws;                                //   752,640 floats
  float* zbuf  = y1 + (size_t)Bsz * C2ch * SS;      // 25,690,112
  float* covdw = zbuf + (size_t)Bsz * Och * SS;     // 77,822,976 (cov, then dw out)
  float* stats = covdw + (size_t)Bsz * CATch * SS;  // 2,048

  // 1) SE scaling of y -> y1
  se_scale_kernel<<<Bsz, 256, 0, stream>>>(y, fc1_w, fc2_w, y1);

  // 2) depthwise conv + per-(b,c) covariance -> cov (in covdw)
  dconv_cov_kernel<<<(Bsz * Cch) / COV_WAVES, COV_WAVES * 32, 0, stream>>>(
      x, dconv_w, dconv_b, covdw);

  // 3) 1x1 conv (GEMM K=1024): z_pre = conv_w @ cov + conv_b -> zbuf
  dim3 g1(7, Och / 64, Bsz);   // 7 N-tiles of 32 cover 196
  wmma_gemm_kernel<<<g1, 256, 0, stream>>>(conv_w, covdw, conv_b, zbuf,
                                           Cch, Och);

  // 4/5) BN (batch stats) + sigmoid -> z (in place in zbuf)
  bn_stats_kernel<<<Och, 256, 0, stream>>>(zbuf, stats, stats + 512);
  bn_apply_kernel<<<(Bsz * Och * SS) / 256, 256, 0, stream>>>(
      zbuf, stats, stats + 512, gamma, beta, /*mode=sigmoid*/ 0);

  // 6) depthwise 3x3 over virtual concat -> dwout (reuses covdw; cov is dead)
  dw_cat_kernel<<<Bsz * CATch, 256, 0, stream>>>(y1, zbuf, x, dw1_w, dw1_b,
                                                 covdw);

  // 7) pointwise conv (GEMM K=1551) -> o_pre directly into d_out
  wmma_gemm_kernel<<<g1, 256, 0, stream>>>(pw1_w, covdw, pw1_b, out,
                                           CATch, Och);

  // 8/9) BN (batch stats) + relu, in place on d_out
  bn_stats_kernel<<<Och, 256, 0, stream>>>(out, stats + 1024, stats + 1536);
  bn_apply_kernel<<<(Bsz * Och * SS) / 256, 256, 0, stream>>>(
      out, stats + 1024, stats + 1536, gamma, beta, /*mode=relu*/ 1);
}